// Hysteresis_33621003993245
// MI455X (gfx1250) — compile-verified
//
#include <hip/hip_runtime.h>

// Preisach hysteresis scan for MI455X (gfx1250, wave32).
//
// Kernel 1: 32 blocks x 256 threads (8 waves). Each thread owns <=5 tril
// cells in registers and walks all T field steps (the serial critical path).
// Per step each lane does only: state update selects + 5 FMAs + one
// conflict-free ds_store_b32 into its wave's 16(t) x 32(lane) LDS tile.
// Every 16 steps, each wave reduces its own tile with 8 chained
// V_WMMA_F32_16X16X4_F32 (ones-column B) -> wave partials, then wave 0
// collapses the 16(t) x 8(wave) matrix with 2 more WMMAs and writes block
// partials to d_ws. Full-FP32 matrix path keeps the 32896-term sum exact-ish.
//
// Kernel 2: sums the 32 block partials per t, applies /tri_size*scale+offset.

typedef __attribute__((ext_vector_type(2))) float v2f;
typedef __attribute__((ext_vector_type(8))) float v8f;

#define NMESH   256
#define TRI     32896           // 256*257/2
#define NTHR    256
#define NWAVE   8
#define NBLK    32
#define MAXCPT  5               // ceil(32896 / (32*256)); threads 0..127 get a 5th cell

__global__ __launch_bounds__(NTHR) void hyst_scan_kernel(
    const float* __restrict__ h,        // applied field, length T (H_MIN=0,H_MAX=1 -> already normalized)
    const float* __restrict__ raw,      // raw density vector, length TRI
    float* __restrict__ partials,       // [NBLK][T] block partial sums
    int T)
{
    __shared__ float tile[NWAVE][16][32];  // [wave][t_local][lane] contributions
    __shared__ float wp2[16][8];           // [t_local][wave] wave partials

    const int tid  = threadIdx.x;
    const int lane = tid & 31;
    const int wave = tid >> 5;
    const int L    = lane & 15;
    const int half = lane >> 4;
    const int gtid = blockIdx.x * NTHR + tid;
    const int stride = NTHR * NBLK;        // 8192 threads total

    // Per-thread private cells: alpha, beta, density, state.
    float al[MAXCPT], be[MAXCPT], dv[MAXCPT], st[MAXCPT];
    #pragma unroll
    for (int c = 0; c < MAXCPT; ++c) {
        int v = gtid + c * stride;
        if (v < TRI) {
            // row j of tril (row-major, diagonal included): j(j+1)/2 <= v
            int j = (int)((sqrtf(8.0f * (float)v + 1.0f) - 1.0f) * 0.5f);
            while ((j + 1) * (j + 2) / 2 <= v) ++j;
            while (j * (j + 1) / 2 > v) --j;
            int k = v - j * (j + 1) / 2;
            al[c] = (float)j * (1.0f / (float)(NMESH - 1));   // alpha threshold (yy)
            be[c] = (float)k * (1.0f / (float)(NMESH - 1));   // beta threshold (xx)
            float r = raw[v];
            dv[c] = (r > 20.0f) ? r : log1pf(expf(r));        // softplus
            st[c] = -1.0f;                                    // init_state = -tril
        } else {
            al[c] = 2.0f; be[c] = -1.0f; dv[c] = 0.0f; st[c] = -1.0f; // inert (dens 0)
        }
    }

    // B matrix for all WMMA reductions: ones in column 0 only.
    // B[k][n] = (n==0) is the same pattern in every K row, so both layout
    // halves get the identical per-lane value.
    v2f bb;
    {
        float b0 = (L == 0) ? 1.0f : 0.0f;
        bb.x = b0; bb.y = b0;
    }

    float hp = 0.0f;  // hs[0] = H_MIN normalized = 0

    for (int tb = 0; tb < T; tb += 16) {
        // ---- 16 serial field steps: compute + one LDS store per lane ----
        for (int tl = 0; tl < 16; ++tl) {
            int t = tb + tl;
            float hc = (t < T) ? h[t] : hp;
            if (hc > hp) {                 // field increasing: alpha < hc -> +1
                #pragma unroll
                for (int c = 0; c < MAXCPT; ++c)
                    if (!(al[c] >= hc)) st[c] = 1.0f;
            } else if (hc < hp) {          // field decreasing: beta > hc -> -1
                #pragma unroll
                for (int c = 0; c < MAXCPT; ++c)
                    if (!(be[c] <= hc)) st[c] = -1.0f;
            }
            hp = hc;

            float contrib = 0.0f;
            #pragma unroll
            for (int c = 0; c < MAXCPT; ++c) contrib += st[c] * dv[c];

            tile[wave][tl][lane] = contrib;    // conflict-free, no cross-lane op
        }
        __syncthreads();   // barrier A (also fences wp2 reuse across batches)

        // ---- per-wave 16x32 lane reduction via 8 chained FP32 WMMAs ----
        // A 16x4 fp32 layout: lanes 0-15 -> M=row, {v0,v1}=K{0,1};
        //                     lanes 16-31 -> {v0,v1}=K{2,3}.
        {
            v8f acc = {};
            #pragma unroll
            for (int c = 0; c < 8; ++c) {
                v2f a = *(const v2f*)&tile[wave][L][c * 4 + half * 2];
                acc = __builtin_amdgcn_wmma_f32_16x16x4_f32(
                          false, a, false, bb, (short)0, acc, false, false);
            }
            // D column 0: lane 0 -> t=0..7 in acc[0..7]; lane 16 -> t=8..15.
            if (L == 0) {
                #pragma unroll
                for (int r = 0; r < 8; ++r)
                    wp2[half * 8 + r][wave] = acc[r];
            }
        }
        __syncthreads();   // barrier B: publish wp2 to wave 0

        // ---- wave 0: 16x8 cross-wave reduction with 2 WMMAs ----
        if (wave == 0) {
            v2f a0, a1;
            a0.x = wp2[L][half ? 2 : 0];
            a0.y = wp2[L][half ? 3 : 1];
            a1.x = wp2[L][half ? 6 : 4];
            a1.y = wp2[L][half ? 7 : 5];

            v8f f = {};
            f = __builtin_amdgcn_wmma_f32_16x16x4_f32(
                    false, a0, false, bb, (short)0, f, false, false);
            f = __builtin_amdgcn_wmma_f32_16x16x4_f32(
                    false, a1, false, bb, (short)0, f, false, false);

            if (L == 0) {
                int tbase = tb + (half ? 8 : 0);
                float* dst = partials + (size_t)blockIdx.x * T + tbase;
                if (tb + 16 <= T) {        // uniform fast path (T%16==0 case)
                    #pragma unroll
                    for (int r = 0; r < 8; ++r) dst[r] = f[r];
                } else {
                    #pragma unroll
                    for (int r = 0; r < 8; ++r)
                        if (tbase + r < T) dst[r] = f[r];
                }
            }
        }
    }
}

__global__ __launch_bounds__(NTHR) void hyst_finalize_kernel(
    const float* __restrict__ partials,
    const float* __restrict__ offset,
    const float* __restrict__ scale,
    float* __restrict__ out,
    int T)
{
    int t = blockIdx.x * blockDim.x + threadIdx.x;
    if (t >= T) return;
    float s = 0.0f;
    #pragma unroll 4
    for (int b = 0; b < NBLK; ++b) s += partials[b * T + t];
    out[t] = s * (1.0f / (float)TRI) * scale[0] + offset[0];
}

extern "C" void kernel_launch(void* const* d_in, const int* in_sizes, int n_in,
                              void* d_out, int out_size, void* d_ws, size_t ws_size,
                              hipStream_t stream) {
    const float* h      = (const float*)d_in[0];   // (T,)
    const float* raw    = (const float*)d_in[1];   // (32896,)
    const float* offset = (const float*)d_in[2];   // (1,)
    const float* scale  = (const float*)d_in[3];   // (1,)
    float* out = (float*)d_out;
    int T = in_sizes[0];

    float* partials = (float*)d_ws;                // NBLK * T floats

    hyst_scan_kernel<<<NBLK, NTHR, 0, stream>>>(h, raw, partials, T);
    hyst_finalize_kernel<<<(T + NTHR - 1) / NTHR, NTHR, 0, stream>>>(
        partials, offset, scale, out, T);
}